// FireGNN_16716012716378
// MI455X (gfx1250) — compile-verified
//
#include <hip/hip_runtime.h>
#include <math.h>

typedef __attribute__((ext_vector_type(2))) float v2f;
typedef __attribute__((ext_vector_type(8))) float v8f;

#define DH 64  // hidden width (out dim of both GCN layers)

// ---------------- degree / normalization ----------------

__global__ void k_deg_init(float* __restrict__ deg, int n) {
  int i = blockIdx.x * blockDim.x + threadIdx.x;
  if (i < n) deg[i] = 1.0f;  // self-loop contributes 1 to every node's degree
}

__global__ void k_deg_edges(const long long* __restrict__ ei, int E,
                            float* __restrict__ deg) {
  int e = blockIdx.x * blockDim.x + threadIdx.x;
  if (e < E) atomicAdd(&deg[(int)ei[(size_t)E + e]], 1.0f);  // dst row
}

__global__ void k_rsqrt_inplace(float* __restrict__ deg, int n) {
  int i = blockIdx.x * blockDim.x + threadIdx.x;
  if (i < n) deg[i] = rsqrtf(deg[i]);  // deg >= 1 always (self-loops)
}

// ---------------- fp32 WMMA GEMM: H[nRows x 64] = X[nRows x K] * W[K x 64] ----------------
// One wave per 16x16 output tile. grid = (ceil(nRows/16), 64/16), block = 32.
// A 16x4 layout: lanes 0-15 hold (M=l, K=k0+0 / k0+1); lanes 16-31 hold (M=l, K=k0+2 / k0+3).
// B 4x16 layout: v0 = row k0 (lanes 0-15) / row k0+2 (lanes 16-31); v1 = rows k0+1 / k0+3.
// C/D 16x16:     element r of lane -> (M = r + 8*(lane>=16), N = lane&15).
template <int K>
__global__ void k_gemm_wmma(const float* __restrict__ X, const float* __restrict__ W,
                            float* __restrict__ H, int nRows) {
  const int m0   = blockIdx.x * 16;
  const int n0   = blockIdx.y * 16;
  const int lane = threadIdx.x;        // 0..31 (wave32)
  const int half = lane >> 4;          // 0 or 1
  const int l    = lane & 15;

  // Clamp A-row for ragged tail tiles so WMMA always runs with full EXEC.
  int mrow = m0 + l;
  if (mrow >= nRows) mrow = nRows - 1;

  v8f acc = {};
  const float* xrow = X + (size_t)mrow * K + half * 2;        // (row, col k0+2*half)
  const float* wrow = W + (size_t)(half * 2) * DH + n0 + l;   // (row k0+2*half, col n0+l)

#pragma unroll
  for (int k0 = 0; k0 < K; k0 += 4) {
    v2f a; a.x = xrow[0]; a.y = xrow[1];          // K = k0+2h, k0+2h+1
    v2f b; b.x = wrow[0]; b.y = wrow[DH];         // rows k0+2h, k0+2h+1 of W
    acc = __builtin_amdgcn_wmma_f32_16x16x4_f32(false, a, false, b,
                                                (short)0, acc, false, false);
    xrow += 4;
    wrow += 4 * DH;
  }

  float* out = H + (size_t)(m0 + half * 8) * DH + n0 + l;
  if (m0 + 16 <= nRows) {
    // full tile (always the case when nRows % 16 == 0): unguarded clause of stores
#pragma unroll
    for (int r = 0; r < 8; ++r) out[(size_t)r * DH] = acc[r];
  } else {
#pragma unroll
    for (int r = 0; r < 8; ++r)
      if (m0 + half * 8 + r < nRows) out[(size_t)r * DH] = acc[r];
  }
}

// ---------------- aggregation ----------------

// agg[i][c] = h[i][c] * dinv[i]^2   (self-loop term; also initializes agg over poison)
__global__ void k_self_init(const float* __restrict__ h, const float* __restrict__ dinv,
                            float* __restrict__ agg, int nel) {
  int idx = blockIdx.x * blockDim.x + threadIdx.x;
  if (idx < nel) {
    float di = dinv[idx >> 6];
    agg[idx] = h[idx] * di * di;
  }
}

// one wave per edge: agg[dst][:] += h[src][:] * dinv[src]*dinv[dst]
__global__ void k_edge_scatter(const long long* __restrict__ ei, int E,
                               const float* __restrict__ h, const float* __restrict__ dinv,
                               float* __restrict__ agg) {
  int e    = blockIdx.x * (blockDim.x >> 5) + (threadIdx.x >> 5);
  int lane = threadIdx.x & 31;
  if (e < E) {
    int s = (int)ei[e];
    int d = (int)ei[(size_t)E + e];
    float norm = dinv[s] * dinv[d];
    const float* hs = h + (size_t)s * DH;
    float* ad = agg + (size_t)d * DH;
    atomicAdd(&ad[lane],      hs[lane]      * norm);
    atomicAdd(&ad[lane + 32], hs[lane + 32] * norm);
  }
}

// act[i][c] = relu(agg[i][c] + b[c])
__global__ void k_bias_relu(const float* __restrict__ agg, const float* __restrict__ b,
                            float* __restrict__ act, int nel) {
  int idx = blockIdx.x * blockDim.x + threadIdx.x;
  if (idx < nel) {
    float v = agg[idx] + b[idx & (DH - 1)];
    act[idx] = v > 0.0f ? v : 0.0f;
  }
}

// ---------------- fused head: sigmoid(relu(agg2 + b2) @ Wo + bo) ----------------
// one wave per node, 2 channels per lane, shfl_xor reduction
__global__ void k_head(const float* __restrict__ agg, const float* __restrict__ b2,
                       const float* __restrict__ Wo, const float* __restrict__ bo,
                       float* __restrict__ out, int n) {
  int node = blockIdx.x * (blockDim.x >> 5) + (threadIdx.x >> 5);
  int lane = threadIdx.x & 31;
  if (node < n) {
    const float* a = agg + (size_t)node * DH;
    float v0 = a[lane]      + b2[lane];      v0 = v0 > 0.0f ? v0 : 0.0f;
    float v1 = a[lane + 32] + b2[lane + 32]; v1 = v1 > 0.0f ? v1 : 0.0f;
    float s = v0 * Wo[lane] + v1 * Wo[lane + 32];
#pragma unroll
    for (int off = 16; off > 0; off >>= 1) s += __shfl_xor(s, off, 32);
    if (lane == 0) out[node] = 1.0f / (1.0f + expf(-(s + bo[0])));
  }
}

// ---------------- launch ----------------

extern "C" void kernel_launch(void* const* d_in, const int* in_sizes, int n_in,
                              void* d_out, int out_size, void* d_ws, size_t ws_size,
                              hipStream_t stream) {
  const float*     x  = (const float*)d_in[0];     // [N,32]
  const long long* ei = (const long long*)d_in[1]; // [2,E] int64
  const float*     W1 = (const float*)d_in[2];     // [32,64]
  const float*     b1 = (const float*)d_in[3];     // [64]
  const float*     W2 = (const float*)d_in[4];     // [64,64]
  const float*     b2 = (const float*)d_in[5];     // [64]
  const float*     Wo = (const float*)d_in[6];     // [64,1]
  const float*     bo = (const float*)d_in[7];     // [1]
  float*           out = (float*)d_out;            // [N]

  const int N = in_sizes[0] / 32;
  const int E = in_sizes[1] / 2;
  const int nel = N * DH;

  // workspace: deg/dinv [N] | bufA [N*64] | bufB [N*64]   (~52 MB)
  float* deg  = (float*)d_ws;
  float* bufA = deg + (((size_t)N + 63) & ~(size_t)63);
  float* bufB = bufA + (size_t)N * DH;

  const int T = 256;
  // normalization coefficients
  k_deg_init     <<<(N + T - 1) / T, T, 0, stream>>>(deg, N);
  k_deg_edges    <<<(E + T - 1) / T, T, 0, stream>>>(ei, E, deg);
  k_rsqrt_inplace<<<(N + T - 1) / T, T, 0, stream>>>(deg, N);

  dim3 ggrid((N + 15) / 16, DH / 16);

  // ---- layer 1: h1 = x @ W1 ; agg ; relu(+b1) ----
  k_gemm_wmma<32><<<ggrid, 32, 0, stream>>>(x, W1, bufA, N);
  k_self_init    <<<(nel + T - 1) / T, T, 0, stream>>>(bufA, deg, bufB, nel);
  k_edge_scatter <<<(E + 7) / 8, 256, 0, stream>>>(ei, E, bufA, deg, bufB);
  k_bias_relu    <<<(nel + T - 1) / T, T, 0, stream>>>(bufB, b1, bufA, nel);

  // ---- layer 2: h2 = act1 @ W2 ; agg ; fused (relu(+b2) @ Wo + bo -> sigmoid) ----
  k_gemm_wmma<64><<<ggrid, 32, 0, stream>>>(bufA, W2, bufB, N);
  k_self_init    <<<(nel + T - 1) / T, T, 0, stream>>>(bufB, deg, bufA, nel);
  k_edge_scatter <<<(E + 7) / 8, 256, 0, stream>>>(ei, E, bufB, deg, bufA);
  k_head         <<<(N + 7) / 8, 256, 0, stream>>>(bufA, b2, Wo, bo, out, N);
}